// GATv2WithGlobal_4389456577272
// MI455X (gfx1250) — compile-verified
//
#include <hip/hip_runtime.h>
#include <hip/hip_bf16.h>
#include <math.h>

#define HC 512
#define HEADS 4
#define CH 128
#define NF 9
#define EF 4
#define GFEAT 50
#define EPSV 1e-5f
#define NSLOPE 0.2f

typedef __attribute__((ext_vector_type(2))) float v2f;
typedef __attribute__((ext_vector_type(8))) float v8f;

// ---------------------------------------------------------------------------
// small helpers
// ---------------------------------------------------------------------------
__device__ __forceinline__ float atomicMaxFloat(float* addr, float value) {
    int* ia = (int*)addr;
    int old = __float_as_int(*addr);
    while (__int_as_float(old) < value) {
        int assumed = old;
        old = atomicCAS(ia, assumed, __float_as_int(value));
        if (old == assumed) break;
    }
    return __int_as_float(old);
}

__global__ void fill_f32_kernel(float* p, float v, int n) {
    int t = blockIdx.x * blockDim.x + threadIdx.x;
    if (t < n) p[t] = v;
}

// ---------------------------------------------------------------------------
// degree + edge-attr sum (per destination node)
// ---------------------------------------------------------------------------
__global__ void deg_ea_kernel(const int* dst, const float* ea,
                              float* deg, float* ea_sum, int E) {
    int j = blockIdx.x * blockDim.x + threadIdx.x;
    if (j >= E) return;
    int d = dst[j];
    atomicAdd(&deg[d], 1.0f);
    #pragma unroll
    for (int f = 0; f < EF; ++f)
        atomicAdd(&ea_sum[(size_t)d * EF + f], ea[(size_t)j * EF + f]);
}

__global__ void ea_mean_kernel(float* ea_sum, const float* deg, int N) {
    int i = blockIdx.x * blockDim.x + threadIdx.x;
    if (i >= N) return;
    float inv = 1.0f / fmaxf(deg[i], 1.0f);
    #pragma unroll
    for (int f = 0; f < EF; ++f) ea_sum[(size_t)i * EF + f] *= inv;
}

// ---------------------------------------------------------------------------
// layer-1 node transforms: xl = x@Wl, xr = x@Wr   (K = 9, scalar)
// ---------------------------------------------------------------------------
__global__ void transform9_kernel(const float* __restrict__ x,
                                  const float* __restrict__ Wl,
                                  const float* __restrict__ Wr,
                                  float* xl, float* xr, int N) {
    size_t t = (size_t)blockIdx.x * blockDim.x + threadIdx.x;
    if (t >= (size_t)N * HC) return;
    int n = (int)(t >> 9);
    int c = (int)(t & (HC - 1));
    const float* xrow = x + (size_t)n * NF;
    float sl = 0.f, sr = 0.f;
    #pragma unroll
    for (int k = 0; k < NF; ++k) {
        float xv = xrow[k];
        sl += xv * Wl[k * HC + c];
        sr += xv * Wr[k * HC + c];
    }
    xl[t] = sl;
    xr[t] = sr;
}

// ---------------------------------------------------------------------------
// WMMA fp32 GEMM:  C[M x 512] = A[M x 512] @ W[512 x 512]
// block = 128 threads (4 waves). Each wave owns a 16x32 output strip
// (2 accumulators sharing one A fragment); block covers 128 columns.
// A tile (16 x 512) staged in LDS with padded stride 516 (bank-conflict free).
// ---------------------------------------------------------------------------
#define LDA 516
__global__ __launch_bounds__(128) void gemm_wmma_kernel(
        const float* __restrict__ A, const float* __restrict__ W,
        float* __restrict__ C, int M) {
    __shared__ float sA[16 * LDA];

    int m0 = blockIdx.x * 16;
    bool full = (m0 + 16 <= M);

    // cooperative load of A tile (16 rows x 512 cols) as float4
    if (full) {
        for (int i = threadIdx.x; i < 16 * 128; i += 128) {
            int r  = i >> 7;
            int c4 = i & 127;
            float4 v = *(const float4*)(A + (size_t)(m0 + r) * HC + c4 * 4);
            float* dstp = &sA[r * LDA + c4 * 4];
            dstp[0] = v.x; dstp[1] = v.y; dstp[2] = v.z; dstp[3] = v.w;
        }
    } else {
        for (int i = threadIdx.x; i < 16 * 128; i += 128) {
            int r  = i >> 7;
            int c4 = i & 127;
            int m  = m0 + r;
            float4 v = make_float4(0.f, 0.f, 0.f, 0.f);
            if (m < M) v = *(const float4*)(A + (size_t)m * HC + c4 * 4);
            float* dstp = &sA[r * LDA + c4 * 4];
            dstp[0] = v.x; dstp[1] = v.y; dstp[2] = v.z; dstp[3] = v.w;
        }
    }
    __syncthreads();

    int wave  = threadIdx.x >> 5;
    int lane  = threadIdx.x & 31;
    int half  = lane >> 4;        // 0: K pair {0,1}, 1: K pair {2,3}
    int l16   = lane & 15;
    int ncol0 = blockIdx.y * 128 + wave * 32;

    v8f acc0 = {};
    v8f acc1 = {};
    for (int kk = 0; kk < HC; kk += 4) {
        int k0 = kk + half * 2;
        // A 16x4 fragment: M = l16, K = k0 + v (shared by both column tiles)
        v2f av;
        av.x = sA[l16 * LDA + k0];
        av.y = sA[l16 * LDA + k0 + 1];
        // B 4x16 fragments: N = l16 (+16), K = k0 + v
        const float* wp = W + (size_t)k0 * HC + ncol0 + l16;
        __builtin_prefetch(wp + 4 * HC, 0, 1);
        v2f b0, b1;
        b0.x = wp[0];
        b0.y = wp[HC];
        b1.x = wp[16];
        b1.y = wp[HC + 16];
        acc0 = __builtin_amdgcn_wmma_f32_16x16x4_f32(
            false, av, false, b0, (short)0, acc0, false, false);
        acc1 = __builtin_amdgcn_wmma_f32_16x16x4_f32(
            false, av, false, b1, (short)0, acc1, false, false);
    }

    // D layout: lanes 0-15 -> M = v, lanes 16-31 -> M = 8 + v; N = l16
    int n    = ncol0 + l16;
    int mrow = m0 + half * 8;
    if (full) {
        #pragma unroll
        for (int v = 0; v < 8; ++v) {
            float* cp = C + (size_t)(mrow + v) * HC + n;
            cp[0]  = acc0[v];
            cp[16] = acc1[v];
        }
    } else {
        #pragma unroll
        for (int v = 0; v < 8; ++v) {
            int m = mrow + v;
            if (m < M) {
                float* cp = C + (size_t)m * HC + n;
                cp[0]  = acc0[v];
                cp[16] = acc1[v];
            }
        }
    }
}

// ---------------------------------------------------------------------------
// edge logits: e = leaky_relu(xl[s]+xr[d]+ee); logit = <e, att[h]>; atomic max
// one wave per (edge or self-loop)
// ---------------------------------------------------------------------------
__global__ __launch_bounds__(256) void edge_logits_kernel(
        const float* __restrict__ xl, const float* __restrict__ xr,
        const int* __restrict__ src, const int* __restrict__ dst,
        const float* __restrict__ edge_attr, const float* __restrict__ ea_mean,
        const float* __restrict__ We, const float* __restrict__ att,
        float* logits, float* node_max, int E, int N) {
    __shared__ float sWe[EF * HC];
    __shared__ float sAtt[HC];
    for (int i = threadIdx.x; i < EF * HC; i += blockDim.x) sWe[i] = We[i];
    for (int i = threadIdx.x; i < HC; i += blockDim.x) sAtt[i] = att[i];
    __syncthreads();

    int wave = threadIdx.x >> 5;
    int lane = threadIdx.x & 31;
    int j = blockIdx.x * (blockDim.x >> 5) + wave;
    int T = E + N;
    if (j >= T) return;

    int s, d;
    const float* ea;
    if (j < E) { s = src[j]; d = dst[j]; ea = edge_attr + (size_t)j * EF; }
    else       { s = d = j - E;          ea = ea_mean + (size_t)(j - E) * EF; }
    float e0 = ea[0], e1 = ea[1], e2 = ea[2], e3 = ea[3];
    const float* pl = xl + (size_t)s * HC;
    const float* pr = xr + (size_t)d * HC;

    #pragma unroll
    for (int h = 0; h < HEADS; ++h) {
        float acc = 0.f;
        int base = h * CH;
        #pragma unroll
        for (int it = 0; it < CH / 32; ++it) {
            int c = base + it * 32 + lane;
            float ee = e0 * sWe[c] + e1 * sWe[HC + c] +
                       e2 * sWe[2 * HC + c] + e3 * sWe[3 * HC + c];
            float v = pl[c] + pr[c] + ee;
            v = v > 0.f ? v : NSLOPE * v;
            acc += sAtt[c] * v;
        }
        #pragma unroll
        for (int off = 16; off > 0; off >>= 1) acc += __shfl_xor(acc, off, 32);
        if (lane == 0) {
            logits[(size_t)j * HEADS + h] = acc;
            atomicMaxFloat(&node_max[(size_t)d * HEADS + h], acc);
        }
    }
}

// ---------------------------------------------------------------------------
// segment softmax: ex = exp(logit - max[d]); sum[d] += ex
// ---------------------------------------------------------------------------
__global__ void edge_softmax_kernel(const float* __restrict__ logits,
                                    const float* __restrict__ node_max,
                                    const int* __restrict__ dst,
                                    float* exbuf, float* node_sum, int E, int N) {
    int T = E + N;
    int t = blockIdx.x * blockDim.x + threadIdx.x;
    if (t >= T * HEADS) return;
    int j = t >> 2;
    int h = t & 3;
    int d = (j < E) ? dst[j] : (j - E);
    float ex = __expf(logits[t] - node_max[(size_t)d * HEADS + h]);
    exbuf[t] = ex;
    atomicAdd(&node_sum[(size_t)d * HEADS + h], ex);
}

// ---------------------------------------------------------------------------
// scatter: out[d] += alpha * xl[s]   (one wave per edge)
// ---------------------------------------------------------------------------
__global__ __launch_bounds__(256) void edge_scatter_kernel(
        const float* __restrict__ xl,
        const int* __restrict__ src, const int* __restrict__ dst,
        const float* __restrict__ exbuf, const float* __restrict__ node_sum,
        float* outb, int E, int N) {
    int wave = threadIdx.x >> 5;
    int lane = threadIdx.x & 31;
    int j = blockIdx.x * (blockDim.x >> 5) + wave;
    int T = E + N;
    if (j >= T) return;
    int s, d;
    if (j < E) { s = src[j]; d = dst[j]; }
    else       { s = d = j - E; }

    float av = 0.f;
    if (lane < HEADS)
        av = exbuf[(size_t)j * HEADS + lane] / node_sum[(size_t)d * HEADS + lane];

    const float* pl = xl + (size_t)s * HC;
    float* po = outb + (size_t)d * HC;
    #pragma unroll
    for (int it = 0; it < HC / 32; ++it) {
        int c = it * 32 + lane;
        float alpha = __shfl(av, c >> 7, 32);
        atomicAdd(&po[c], alpha * pl[c]);
    }
}

// ---------------------------------------------------------------------------
// bias + batch-norm + relu (in place)
// ---------------------------------------------------------------------------
__global__ void bn_relu_kernel(float* h, const float* __restrict__ b,
                               const float* __restrict__ gamma,
                               const float* __restrict__ beta,
                               const float* __restrict__ mean,
                               const float* __restrict__ var, int N) {
    size_t t = (size_t)blockIdx.x * blockDim.x + threadIdx.x;
    if (t >= (size_t)N * HC) return;
    int c = (int)(t & (HC - 1));
    float v = h[t] + b[c];
    v = (v - mean[c]) * rsqrtf(var[c] + EPSV) * gamma[c] + beta[c];
    h[t] = fmaxf(v, 0.f);
}

// ---------------------------------------------------------------------------
// pooling
// ---------------------------------------------------------------------------
__global__ void cnt_kernel(const int* batch, float* cnt, int N) {
    int i = blockIdx.x * blockDim.x + threadIdx.x;
    if (i < N) atomicAdd(&cnt[batch[i]], 1.0f);
}

__global__ void pool_kernel(const float* __restrict__ h,
                            const int* __restrict__ batch,
                            float* pooled, int N) {
    size_t t = (size_t)blockIdx.x * blockDim.x + threadIdx.x;
    if (t >= (size_t)N * HC) return;
    int n = (int)(t >> 9);
    int c = (int)(t & (HC - 1));
    atomicAdd(&pooled[(size_t)batch[n] * HC + c], h[t]);
}

// ---------------------------------------------------------------------------
// MLP head: z = relu([pooled/cnt, gfeat] @ fc1 + b); out = z @ fc2 + b
// one block (128 threads) per graph
// ---------------------------------------------------------------------------
__global__ __launch_bounds__(128) void head_kernel(
        const float* __restrict__ pooled, const float* __restrict__ cnt,
        const float* __restrict__ gfeat,
        const float* __restrict__ fc1_w, const float* __restrict__ fc1_b,
        const float* __restrict__ fc2_w, const float* __restrict__ fc2_b,
        float* out, int G) {
    __shared__ float zbuf[CH];
    int g = blockIdx.x;
    if (g >= G) return;
    int c = threadIdx.x;
    float inv = 1.0f / fmaxf(cnt[g], 1.0f);
    float acc = fc1_b[c];
    const float* pr = pooled + (size_t)g * HC;
    for (int k = 0; k < HC; ++k)
        acc += pr[k] * inv * fc1_w[(size_t)k * CH + c];
    const float* gr = gfeat + (size_t)g * GFEAT;
    for (int k = 0; k < GFEAT; ++k)
        acc += gr[k] * fc1_w[(size_t)(HC + k) * CH + c];
    zbuf[c] = fmaxf(acc, 0.f) * fc2_w[c];
    __syncthreads();
    for (int sred = CH / 2; sred > 0; sred >>= 1) {
        if (c < sred) zbuf[c] += zbuf[c + sred];
        __syncthreads();
    }
    if (c == 0) out[g] = zbuf[0] + fc2_b[0];
}

// ---------------------------------------------------------------------------
// launcher
// ---------------------------------------------------------------------------
extern "C" void kernel_launch(void* const* d_in, const int* in_sizes, int n_in,
                              void* d_out, int out_size, void* d_ws, size_t ws_size,
                              hipStream_t stream) {
    const float* x        = (const float*)d_in[0];
    const float* eattr    = (const float*)d_in[1];
    const float* gfeat    = (const float*)d_in[2];
    const float* Wl1      = (const float*)d_in[3];
    const float* Wr1      = (const float*)d_in[4];
    const float* We1      = (const float*)d_in[5];
    const float* att1     = (const float*)d_in[6];
    const float* b1       = (const float*)d_in[7];
    const float* gamma1   = (const float*)d_in[8];
    const float* beta1    = (const float*)d_in[9];
    const float* mean1    = (const float*)d_in[10];
    const float* var1     = (const float*)d_in[11];
    const float* Wl2      = (const float*)d_in[12];
    const float* Wr2      = (const float*)d_in[13];
    const float* We2      = (const float*)d_in[14];
    const float* att2     = (const float*)d_in[15];
    const float* b2       = (const float*)d_in[16];
    const float* gamma2   = (const float*)d_in[17];
    const float* beta2    = (const float*)d_in[18];
    const float* mean2    = (const float*)d_in[19];
    const float* var2     = (const float*)d_in[20];
    const float* fc1_w    = (const float*)d_in[21];
    const float* fc1_b    = (const float*)d_in[22];
    const float* fc2_w    = (const float*)d_in[23];
    const float* fc2_b    = (const float*)d_in[24];
    const int*   eidx     = (const int*)d_in[25];
    const int*   batch    = (const int*)d_in[26];

    const int N = in_sizes[0] / NF;
    const int E = in_sizes[25] / 2;
    const int G = in_sizes[2] / GFEAT;
    const int T = E + N;
    const int* src = eidx;
    const int* dst = eidx + E;

    // workspace carve-up
    char* ws = (char*)d_ws;
    size_t off = 0;
    auto carve = [&](size_t bytes) {
        float* p = (float*)(ws + off);
        off += (bytes + 255) & ~(size_t)255;
        return p;
    };
    const size_t NB = (size_t)N * HC * sizeof(float);
    float* bufA     = carve(NB);                         // xl
    float* bufB     = carve(NB);                         // xr
    float* bufC     = carve(NB);                         // layer-1 agg -> h1
    float* bufD     = carve(NB);                         // layer-2 agg -> h2
    float* deg      = carve((size_t)N * sizeof(float));
    float* ea_mean  = carve((size_t)N * EF * sizeof(float));
    float* logits   = carve((size_t)T * HEADS * sizeof(float));
    float* exbuf    = carve((size_t)T * HEADS * sizeof(float));
    float* node_max = carve((size_t)N * HEADS * sizeof(float));
    float* node_sum = carve((size_t)N * HEADS * sizeof(float));
    float* pooled   = carve((size_t)G * HC * sizeof(float));
    float* cnt      = carve((size_t)G * sizeof(float));
    (void)ws_size;

    const int TB = 256;
    dim3 gemm_grid((N + 15) / 16, HC / 128);

    // ---- shared preprocessing -------------------------------------------
    hipMemsetAsync(deg, 0, (size_t)N * sizeof(float), stream);
    hipMemsetAsync(ea_mean, 0, (size_t)N * EF * sizeof(float), stream);
    deg_ea_kernel<<<(E + TB - 1) / TB, TB, 0, stream>>>(dst, eattr, deg, ea_mean, E);
    ea_mean_kernel<<<(N + TB - 1) / TB, TB, 0, stream>>>(ea_mean, deg, N);

    // ---- layer 1 ---------------------------------------------------------
    transform9_kernel<<<(int)(((size_t)N * HC + TB - 1) / TB), TB, 0, stream>>>(
        x, Wl1, Wr1, bufA, bufB, N);
    fill_f32_kernel<<<(N * HEADS + TB - 1) / TB, TB, 0, stream>>>(node_max, -INFINITY, N * HEADS);
    hipMemsetAsync(node_sum, 0, (size_t)N * HEADS * sizeof(float), stream);
    hipMemsetAsync(bufC, 0, NB, stream);
    edge_logits_kernel<<<(T + 7) / 8, 256, 0, stream>>>(
        bufA, bufB, src, dst, eattr, ea_mean, We1, att1, logits, node_max, E, N);
    edge_softmax_kernel<<<(T * HEADS + TB - 1) / TB, TB, 0, stream>>>(
        logits, node_max, dst, exbuf, node_sum, E, N);
    edge_scatter_kernel<<<(T + 7) / 8, 256, 0, stream>>>(
        bufA, src, dst, exbuf, node_sum, bufC, E, N);
    bn_relu_kernel<<<(int)(((size_t)N * HC + TB - 1) / TB), TB, 0, stream>>>(
        bufC, b1, gamma1, beta1, mean1, var1, N);

    // ---- layer 2 (WMMA transforms) --------------------------------------
    gemm_wmma_kernel<<<gemm_grid, 128, 0, stream>>>(bufC, Wl2, bufA, N);
    gemm_wmma_kernel<<<gemm_grid, 128, 0, stream>>>(bufC, Wr2, bufB, N);
    fill_f32_kernel<<<(N * HEADS + TB - 1) / TB, TB, 0, stream>>>(node_max, -INFINITY, N * HEADS);
    hipMemsetAsync(node_sum, 0, (size_t)N * HEADS * sizeof(float), stream);
    hipMemsetAsync(bufD, 0, NB, stream);
    edge_logits_kernel<<<(T + 7) / 8, 256, 0, stream>>>(
        bufA, bufB, src, dst, eattr, ea_mean, We2, att2, logits, node_max, E, N);
    edge_softmax_kernel<<<(T * HEADS + TB - 1) / TB, TB, 0, stream>>>(
        logits, node_max, dst, exbuf, node_sum, E, N);
    edge_scatter_kernel<<<(T + 7) / 8, 256, 0, stream>>>(
        bufA, src, dst, exbuf, node_sum, bufD, E, N);
    bn_relu_kernel<<<(int)(((size_t)N * HC + TB - 1) / TB), TB, 0, stream>>>(
        bufD, b2, gamma2, beta2, mean2, var2, N);

    // ---- pooling + head --------------------------------------------------
    hipMemsetAsync(pooled, 0, (size_t)G * HC * sizeof(float), stream);
    hipMemsetAsync(cnt, 0, (size_t)G * sizeof(float), stream);
    cnt_kernel<<<(N + TB - 1) / TB, TB, 0, stream>>>(batch, cnt, N);
    pool_kernel<<<(int)(((size_t)N * HC + TB - 1) / TB), TB, 0, stream>>>(
        bufD, batch, pooled, N);
    head_kernel<<<G, CH, 0, stream>>>(pooled, cnt, gfeat, fc1_w, fc1_b,
                                      fc2_w, fc2_b, (float*)d_out, G);
}